// ortho_loss_42434276885043
// MI455X (gfx1250) — compile-verified
//
#include <hip/hip_runtime.h>

typedef __attribute__((ext_vector_type(2))) float v2f;
typedef __attribute__((ext_vector_type(4))) float v4f;
typedef __attribute__((ext_vector_type(8))) float v8f;

#define NSYS 16
#define D 64
#define WS_FLOATS (NSYS * D + NSYS)   // 1024 sums + 16 counts

// ---------------------------------------------------------------------------
// Kernel 0: zero the accumulation workspace (must run every launch; harness
// does not re-poison between graph replays).
// ---------------------------------------------------------------------------
__global__ __launch_bounds__(256) void zero_ws_kernel(float* __restrict__ ws) {
    int i = blockIdx.x * blockDim.x + threadIdx.x;
    if (i < WS_FLOATS) ws[i] = 0.0f;
}

// ---------------------------------------------------------------------------
// Kernel 1: streaming pass. 16 lanes per row, float4 per lane (one 256B row
// per 16-lane group, fully coalesced). Norm via shfl_xor over the 16-lane
// group, then LDS f32 atomics into per-block [16][64] accumulators, flushed
// once per block via global f32 atomics.
// ---------------------------------------------------------------------------
__global__ __launch_bounds__(256) void accum_kernel(const float* __restrict__ x,
                                                    const int* __restrict__ sys,
                                                    float* __restrict__ ws,
                                                    int N) {
    __shared__ float sacc[NSYS * D];
    __shared__ float scnt[NSYS];

    const int tid = threadIdx.x;
    for (int i = tid; i < NSYS * D; i += 256) sacc[i] = 0.0f;
    if (tid < NSYS) scnt[tid] = 0.0f;
    __syncthreads();

    const int lane16 = tid & 15;       // which float4 of the row
    const int rowgrp = tid >> 4;       // 0..15 row slot within block
    const int stride = gridDim.x * 16;

    for (int r = blockIdx.x * 16 + rowgrp; r < N; r += stride) {
        const v4f* p = (const v4f*)(x + (size_t)r * D) + lane16;
        v4f v = __builtin_nontemporal_load(p);   // touch-once stream, keep L2 clean

        float ss = v.x * v.x + v.y * v.y + v.z * v.z + v.w * v.w;
        ss += __shfl_xor(ss, 1, 16);
        ss += __shfl_xor(ss, 2, 16);
        ss += __shfl_xor(ss, 4, 16);
        ss += __shfl_xor(ss, 8, 16);

        float inv = 1.0f / fmaxf(sqrtf(ss), 1e-12f);   // F.normalize eps clamp
        int s = sys[r];

        float* a = &sacc[s * D + lane16 * 4];
        atomicAdd(a + 0, v.x * inv);    // ds_add_f32
        atomicAdd(a + 1, v.y * inv);
        atomicAdd(a + 2, v.z * inv);
        atomicAdd(a + 3, v.w * inv);
        if (lane16 == 0) atomicAdd(&scnt[s], 1.0f);
    }

    __syncthreads();
    for (int i = tid; i < NSYS * D; i += 256) atomicAdd(&ws[i], sacc[i]);
    if (tid < NSYS) atomicAdd(&ws[NSYS * D + tid], scnt[tid]);
}

// ---------------------------------------------------------------------------
// Kernel 2: single wave32. Build means[16][64] in LDS, then Gram matrix
// means @ means^T via 16x V_WMMA_F32_16X16X4_F32 (full f32 accuracy).
// A (16x4, 32-bit) layout:  lanes 0-15: v0=K0,v1=K1 ; lanes 16-31: v0=K2,v1=K3
// B (4x16) mirrors A with M->N, and B = A^T here, so the same register pair
// serves as both operands.
// C/D layout: VGPR r = row r (lanes 0-15) / row r+8 (lanes 16-31), col = lane&15.
// ---------------------------------------------------------------------------
__global__ __launch_bounds__(32) void finalize_kernel(const float* __restrict__ ws,
                                                      float* __restrict__ out) {
    __shared__ float means[NSYS * D];

    const int tid = threadIdx.x;   // 0..31, single wave, EXEC all ones
    for (int i = tid; i < NSYS * D; i += 32) {
        float cnt = ws[NSYS * D + (i >> 6)];
        means[i] = ws[i] / cnt;
    }
    __syncthreads();

    const int l = tid & 15;
    const int off = (tid < 16) ? 0 : 2;

    v8f c = {};
    for (int k0 = 0; k0 < D; k0 += 4) {
        v2f a;
        a.x = means[l * D + k0 + off];
        a.y = means[l * D + k0 + off + 1];
        // D = A*B + C,  A = means (16xK), B = means^T (Kx16); symmetric -> a==b
        c = __builtin_amdgcn_wmma_f32_16x16x4_f32(
                /*neg_a=*/false, a, /*neg_b=*/false, a,
                /*c_mod=*/(short)0, c, /*reuse_a=*/false, /*reuse_b=*/false);
    }

    const int col = l;
    const int rbase = (tid < 16) ? 0 : 8;
    for (int r = 0; r < 8; ++r) {
        int row = rbase + r;
        float dot = c[r];
        float dist = (row == col) ? 0.0f : (1.0f - (dot + 1.0f) * 0.5f);
        out[row * NSYS + col] = dist;
    }
}

// ---------------------------------------------------------------------------
extern "C" void kernel_launch(void* const* d_in, const int* in_sizes, int n_in,
                              void* d_out, int out_size, void* d_ws, size_t ws_size,
                              hipStream_t stream) {
    const float* x  = (const float*)d_in[0];   // [N, 64] f32
    const int* sys  = (const int*)d_in[1];     // [N] i32
    const int N     = in_sizes[1];
    float* ws       = (float*)d_ws;            // 1040 floats
    float* out      = (float*)d_out;           // [16,16] f32

    zero_ws_kernel<<<(WS_FLOATS + 255) / 256, 256, 0, stream>>>(ws);

    int blocks = (N + 15) / 16;
    if (blocks > 2048) blocks = 2048;          // ~61 row-iterations per block
    accum_kernel<<<blocks, 256, 0, stream>>>(x, sys, ws, N);

    finalize_kernel<<<1, 32, 0, stream>>>(ws, out);
}